// CritiGraph_70720931496371
// MI455X (gfx1250) — compile-verified
//
#include <hip/hip_runtime.h>
#include <hip/hip_bf16.h>
#include <stdint.h>

// Problem constants (from reference): H=16, TP=8, N=2^16, K=8, C=257, T=S=128
#define H_BITS 16
#define TP 8
#define T_DIM 128
#define S_DIM 128
#define C_DIM 257
#define SB 16                      // s-rows per workgroup
#define BLK 256                    // threads per block (8 waves, wave32)
#define ELEMS (C_DIM * TP)         // 2056 outputs per (t,s)
#define QUADS (ELEMS / 4)          // 514 float4 quads per (t,s)

// 16-byte chunk counts for async staging
#define CH_CNC (ELEMS / 4)         // 514
#define CH_POS (SB * TP / 4)       // 32
#define CH_STA (TP / 4)            // 2
#define CH_NRM (SB / 4)            // 4
#define CH_TOT (CH_CNC + CH_POS + CH_STA + CH_NRM)  // 552

// Truncate a generic pointer to a __shared__ object to its 32-bit LDS offset.
__device__ __forceinline__ unsigned lds_off(const void* p) {
    return (unsigned)(uintptr_t)p;
}

// CDNA5 async global->LDS copy of 16 bytes (GV addressing, tracked by ASYNCcnt).
__device__ __forceinline__ void async_ld16(unsigned lds_addr, const void* gptr) {
    asm volatile("global_load_async_to_lds_b128 %0, %1, off"
                 :
                 : "v"(lds_addr), "v"((unsigned long long)(uintptr_t)gptr)
                 : "memory");
}

// Integer critbit value: sg * (16 - e), where e = frexp-exponent of (|a|^|b|)+1
// = 32 - clz(x).  Reference float value is (this)/16 * norm.
__device__ __forceinline__ int critv(int a, int b) {
    int ua = a < 0 ? -a : a;
    int ub = b < 0 ? -b : b;
    unsigned x = (unsigned)(ua ^ ub) + 1u;       // in [1, 65536]
    int iv = __clz((int)x) - H_BITS;             // clz in [15,31] -> iv in [-1,15]
    return ((a ^ b) < 0) ? -iv : iv;             // sign product
}

__device__ __forceinline__ float4 quad_val(int4 q, int4 p4, int4 a4, int sm, float nf) {
    float4 r;
    r.x = (float)(sm - a4.x + critv(q.x, p4.x)) * nf;
    r.y = (float)(sm - a4.y + critv(q.y, p4.y)) * nf;
    r.z = (float)(sm - a4.z + critv(q.z, p4.z)) * nf;
    r.w = (float)(sm - a4.w + critv(q.w, p4.w)) * nf;
    return r;
}

__global__ __launch_bounds__(BLK) void critigraph_kernel(
    const int*   __restrict__ sta,   // (T, TP)
    const int*   __restrict__ pos,   // (T, S, TP)
    const int*   __restrict__ cnc,   // (T, C, TP)
    const float* __restrict__ nrm,   // (T, S)
    float*       __restrict__ out)   // (T, S, C, TP)
{
    __shared__ __align__(16) int   cnc_sh[ELEMS];     // 8224 B, reused for SB s-rows
    __shared__ __align__(16) int   pos_sh[SB * TP];   // 512 B
    __shared__ __align__(16) int   sta_sh[TP];        // 32 B
    __shared__ __align__(16) float nrm_sh[SB];        // 64 B
    __shared__ __align__(16) int   A_sh[SB * TP];     // integer critv(sta,pos)
    __shared__ int sum_sh[SB];                        // row sums of A

    const int t   = blockIdx.y;
    const int s0  = blockIdx.x * SB;
    const int tid = (int)threadIdx.x;

    const int*   cnc_t = cnc + (size_t)t * ELEMS;
    const int*   pos_t = pos + ((size_t)t * S_DIM + s0) * TP;
    const int*   sta_t = sta + (size_t)t * TP;
    const float* nrm_t = nrm + (size_t)t * S_DIM + s0;

    // ---- Stage all per-(t, s-block) inputs into LDS with async tensor-path copies.
    // Chunk map: [0,514) cnc | [514,546) pos | [546,548) sta | [548,552) norm
    for (int i = tid; i < CH_TOT; i += BLK) {
        unsigned dst;
        const void* src;
        if (i < CH_CNC) {
            dst = lds_off(&cnc_sh[i * 4]);              src = cnc_t + i * 4;
        } else if (i < CH_CNC + CH_POS) {
            int j = i - CH_CNC;
            dst = lds_off(&pos_sh[j * 4]);              src = pos_t + j * 4;
        } else if (i < CH_CNC + CH_POS + CH_STA) {
            int j = i - (CH_CNC + CH_POS);
            dst = lds_off(&sta_sh[j * 4]);              src = sta_t + j * 4;
        } else {
            int j = i - (CH_CNC + CH_POS + CH_STA);
            dst = lds_off(&nrm_sh[j * 4]);              src = nrm_t + j * 4;
        }
        async_ld16(dst, src);
    }
    asm volatile("s_wait_asynccnt 0x0" ::: "memory");
    __syncthreads();

    // ---- A[si][tp] = critv(sta[tp], pos[si][tp]) and its row sums (integer form).
    if (tid < SB * TP) {
        int si = tid >> 3, tp = tid & 7;
        A_sh[tid] = critv(sta_sh[tp], pos_sh[si * TP + tp]);
    }
    __syncthreads();
    if (tid < SB) {
        int s = 0;
        #pragma unroll
        for (int j = 0; j < TP; ++j) s += A_sh[tid * TP + j];
        sum_sh[tid] = s;
    }
    __syncthreads();

    // ---- Each thread owns quads {tid, tid+256} (+ remainder quads 512,513 on tid<2).
    // Element base of quad q is 4q; tp-half h4 = (4q)&7 is identical for all its quads.
    const int h4 = (tid & 1) * 4;
    const int4 q0 = *(const int4*)&cnc_sh[tid * 4];
    const int4 q1 = *(const int4*)&cnc_sh[(tid + BLK) * 4];
    int4 q2 = make_int4(0, 0, 0, 0);
    if (tid < 2) q2 = *(const int4*)&cnc_sh[(2 * BLK + tid) * 4];

    for (int si = 0; si < SB; ++si) {
        const int4  p4 = *(const int4*)&pos_sh[si * TP + h4];
        const int4  a4 = *(const int4*)&A_sh[si * TP + h4];
        const int   sm = sum_sh[si];
        const float nf = nrm_sh[si] * 0.0078125f;   // norm / (16 * TP)
        float* ob = out + ((size_t)(t * S_DIM + s0 + si)) * ELEMS;

        *(float4*)(ob + tid * 4)        = quad_val(q0, p4, a4, sm, nf);
        *(float4*)(ob + 1024 + tid * 4) = quad_val(q1, p4, a4, sm, nf);
        if (tid < 2)
            *(float4*)(ob + 2048 + tid * 4) = quad_val(q2, p4, a4, sm, nf);
    }
}

extern "C" void kernel_launch(void* const* d_in, const int* in_sizes, int n_in,
                              void* d_out, int out_size, void* d_ws, size_t ws_size,
                              hipStream_t stream) {
    (void)in_sizes; (void)n_in; (void)out_size; (void)d_ws; (void)ws_size;
    const int*   sta = (const int*)d_in[0];   // (T, TP)
    const int*   pos = (const int*)d_in[1];   // (T, S, TP)
    const int*   cnc = (const int*)d_in[2];   // (T, C, TP)
    const float* nrm = (const float*)d_in[3]; // (T, S)
    float* out = (float*)d_out;               // (T, S, C, TP)

    dim3 grid(S_DIM / SB, T_DIM);             // 8 x 128 = 1024 workgroups
    critigraph_kernel<<<grid, BLK, 0, stream>>>(sta, pos, cnc, nrm, out);
}